// conv3d_56392920596825
// MI455X (gfx1250) — compile-verified
//
#include <hip/hip_runtime.h>

typedef __attribute__((ext_vector_type(2))) float v2f;
typedef __attribute__((ext_vector_type(4))) float v4f;
typedef __attribute__((ext_vector_type(8))) float v8f;

#define KVOL   27
#define MPAIR  100000
#define NOUT   200000
#define CIN    64
#define COUT   64
#define TILE_M 128
#define LSTR   68   // padded LDS row stride (floats): 272B/row -> 16B aligned, bank step 4

__global__ __launch_bounds__(256) void bias_init_kernel(
    const float* __restrict__ bias, float* __restrict__ out, int total)
{
    int i = blockIdx.x * blockDim.x + threadIdx.x;
    if (i < total) out[i] = bias[i & (COUT - 1)];
}

__global__ __launch_bounds__(256) void sparse_conv_wmma_kernel(
    const float* __restrict__ in_feats,   // [N_IN, CIN]
    const float* __restrict__ wk,         // [KVOL, CIN, COUT]
    const int*   __restrict__ imap,       // [KVOL, MPAIR]
    const int*   __restrict__ omap,       // [KVOL, MPAIR]
    float*       __restrict__ out)        // [NOUT, COUT]
{
    __shared__ float sA[TILE_M * LSTR];    // gathered input tile, padded rows
    __shared__ float sWt[COUT * LSTR];     // weights transposed: Wt[o][c] = W[c][o]
    __shared__ int   sOmap[TILE_M];

    const int k    = blockIdx.y;
    const int m0   = blockIdx.x * TILE_M;
    const int tid  = threadIdx.x;
    const int lane = tid & 31;
    const int wave = tid >> 5;

    // ---- stage omap rows ----
    if (tid < TILE_M) {
        int m = m0 + tid;
        sOmap[tid] = (m < MPAIR) ? omap[k * MPAIR + m] : -1;
    }

    // ---- stage W_k transposed into LDS (64x64 f32 = 16KB) ----
    {
        const float* W = wk + (size_t)k * (CIN * COUT);
        #pragma unroll
        for (int i = 0; i < (CIN * COUT / 4) / 256; ++i) {
            int idx = tid + i * 256;
            int c   = idx >> 4;           // input channel (row of W)
            int o4  = (idx & 15) << 2;    // output channel group of 4
            v4f w = *(const v4f*)(W + c * COUT + o4);
            sWt[(o4 + 0) * LSTR + c] = w.x;
            sWt[(o4 + 1) * LSTR + c] = w.y;
            sWt[(o4 + 2) * LSTR + c] = w.z;
            sWt[(o4 + 3) * LSTR + c] = w.w;
        }
    }

    // ---- gather A tile: 128 rows x 64 ch (32KB), float4 per chunk ----
    {
        #pragma unroll
        for (int i = 0; i < (TILE_M * (CIN / 4)) / 256; ++i) {
            int idx = tid + i * 256;
            int r   = idx >> 4;           // row within tile
            int c4  = (idx & 15) << 2;    // channel group of 4
            int m   = m0 + r;
            v4f a = {0.f, 0.f, 0.f, 0.f};
            if (m < MPAIR) {
                int src = imap[k * MPAIR + m];
                a = *(const v4f*)(in_feats + (size_t)src * CIN + c4);
            }
            *(v4f*)(&sA[r * LSTR + c4]) = a;   // (r*68 + c4)*4 is 16B aligned
        }
    }

    __syncthreads();

    // ---- WMMA compute: wave owns rows [rbase, rbase+16), all 64 out channels ----
    const int rbase = wave << 4;
    const int col   = lane & 15;     // A: row-in-subtile; B/C/D: column (N)
    const int kh    = lane >> 4;     // K-half selector for A/B fragments

    v8f acc0 = {}, acc1 = {}, acc2 = {}, acc3 = {};

    // A frag: lane -> row = lane%16, floats {A[row][kb+2*kh], A[row][kb+2*kh+1]}
    const float* aRow = &sA[(rbase + col) * LSTR + 2 * kh];
    // B frag: lane -> col n, floats {W[kb+2*kh][n], W[kb+2*kh+1][n]} = contiguous in Wt
    const float* b0p = &sWt[(col +  0) * LSTR + 2 * kh];
    const float* b1p = &sWt[(col + 16) * LSTR + 2 * kh];
    const float* b2p = &sWt[(col + 32) * LSTR + 2 * kh];
    const float* b3p = &sWt[(col + 48) * LSTR + 2 * kh];

    #pragma unroll
    for (int kb = 0; kb < CIN; kb += 4) {
        v2f a  = *(const v2f*)(aRow + kb);
        v2f b0 = *(const v2f*)(b0p + kb);
        v2f b1 = *(const v2f*)(b1p + kb);
        v2f b2 = *(const v2f*)(b2p + kb);
        v2f b3 = *(const v2f*)(b3p + kb);
        acc0 = __builtin_amdgcn_wmma_f32_16x16x4_f32(false, a, false, b0, (short)0, acc0, false, false);
        acc1 = __builtin_amdgcn_wmma_f32_16x16x4_f32(false, a, false, b1, (short)0, acc1, false, false);
        acc2 = __builtin_amdgcn_wmma_f32_16x16x4_f32(false, a, false, b2, (short)0, acc2, false, false);
        acc3 = __builtin_amdgcn_wmma_f32_16x16x4_f32(false, a, false, b3, (short)0, acc3, false, false);
    }

    // ---- scatter-add: C/D layout -> lane holds (M = v + 8*kh, N = col + 16*ntile) ----
    #pragma unroll
    for (int v = 0; v < 8; ++v) {
        int r    = rbase + v + (kh << 3);
        int orow = sOmap[r];
        if (orow >= 0) {
            float* dst = out + (size_t)orow * COUT + col;
            atomicAdd(dst +  0, acc0[v]);
            atomicAdd(dst + 16, acc1[v]);
            atomicAdd(dst + 32, acc2[v]);
            atomicAdd(dst + 48, acc3[v]);
        }
    }
}

extern "C" void kernel_launch(void* const* d_in, const int* in_sizes, int n_in,
                              void* d_out, int out_size, void* d_ws, size_t ws_size,
                              hipStream_t stream)
{
    const float* in_feats = (const float*)d_in[0];
    const float* wk       = (const float*)d_in[1];
    const float* bias     = (const float*)d_in[2];
    const int*   imap     = (const int*)d_in[3];
    const int*   omap     = (const int*)d_in[4];
    float*       out      = (float*)d_out;

    // 1) out[n][c] = bias[c]
    int total = NOUT * COUT;
    bias_init_kernel<<<(total + 255) / 256, 256, 0, stream>>>(bias, out, total);

    // 2) gather -> WMMA GEMM -> atomic scatter-add, per kernel offset
    dim3 grid((MPAIR + TILE_M - 1) / TILE_M, KVOL);
    sparse_conv_wmma_kernel<<<grid, 256, 0, stream>>>(in_feats, wk, imap, omap, out);
}